// Attention_84172769068272
// MI455X (gfx1250) — compile-verified
//
#include <hip/hip_runtime.h>

typedef float v2f __attribute__((ext_vector_type(2)));
typedef float v8f __attribute__((ext_vector_type(8)));
typedef unsigned int v4u __attribute__((ext_vector_type(4)));
typedef int v4i __attribute__((ext_vector_type(4)));
typedef int v8i __attribute__((ext_vector_type(8)));

constexpr int kB = 64, kT = 1380, kX = 96, kH = 192, kC = 345;
constexpr int kMT = (kC + 15) / 16;   // 22 m-tiles over C rows

#if defined(__has_builtin)
#  if __has_builtin(__builtin_amdgcn_tensor_load_to_lds)
#    define HAVE_TDM 1
#  endif
#endif
#ifndef HAVE_TDM
#  define HAVE_TDM 0
#endif

__device__ __forceinline__ v8f wmma_k4(v2f a, v2f b, v8f c) {
  // V_WMMA_F32_16X16X4_F32: fp32 A(16x4) x B(4x16) + C(16x16 f32)
  return __builtin_amdgcn_wmma_f32_16x16x4_f32(false, a, false, b, (short)0, c,
                                               false, false);
}

// ---------------- Stage 1: Z[b] = W2 @ X[b]   (345x1380 . 1380x96) ----------
__global__ __launch_bounds__(32) void k_z(const float* __restrict__ W2,
                                          const float* __restrict__ Xin,
                                          float* __restrict__ Z) {
  const int tile = blockIdx.x;            // 22 * 6
  const int mt = tile / (kX / 16);
  const int nt = tile % (kX / 16);
  const int b = blockIdx.y;
  const int lane = threadIdx.x;
  const int hi = lane >> 4, n = lane & 15;
  const int am = min(mt * 16 + n, kC - 1);        // clamp A row (tail tile)
  const float* __restrict__ Xb = Xin + (size_t)b * kT * kX;
  const float* __restrict__ arow = W2 + (size_t)am * kT;

  v8f acc = {};
  #pragma unroll 5
  for (int k0 = 0; k0 < kT; k0 += 4) {
    const int k = k0 + 2 * hi;
    v2f a = *(const v2f*)(arow + k);
    v2f bb;
    bb.x = Xb[(size_t)k * kX + nt * 16 + n];
    bb.y = Xb[(size_t)(k + 1) * kX + nt * 16 + n];
    acc = wmma_k4(a, bb, acc);
  }
  float* __restrict__ Zb = Z + (size_t)b * kC * kX;
  #pragma unroll
  for (int v = 0; v < 8; ++v) {
    const int row = mt * 16 + v + 8 * hi;
    if (row < kC) Zb[(size_t)row * kX + nt * 16 + n] = acc[v];
  }
}

// ---------------- Stage 2: Y[b] = SCALE * Z[b] @ W1  (345x96 . 96x192) ------
__global__ __launch_bounds__(32) void k_y(const float* __restrict__ Z,
                                          const float* __restrict__ W1,
                                          float* __restrict__ Y) {
  const int tile = blockIdx.x;            // 22 * 12
  const int mt = tile / (kH / 16);
  const int nt = tile % (kH / 16);
  const int b = blockIdx.y;
  const int lane = threadIdx.x;
  const int hi = lane >> 4, n = lane & 15;
  const int am = min(mt * 16 + n, kC - 1);
  const float* __restrict__ Zb = Z + (size_t)b * kC * kX;
  const float* __restrict__ arow = Zb + (size_t)am * kX;

  v8f acc = {};
  #pragma unroll
  for (int k0 = 0; k0 < kX; k0 += 4) {
    const int k = k0 + 2 * hi;
    v2f a = *(const v2f*)(arow + k);
    v2f bb;
    bb.x = W1[(size_t)k * kH + nt * 16 + n];
    bb.y = W1[(size_t)(k + 1) * kH + nt * 16 + n];
    acc = wmma_k4(a, bb, acc);
  }
  const float scale = 7.3656956e-3f;      // 1/sqrt(96*192), folded into Y
  float* __restrict__ Yb = Y + (size_t)b * kC * kH;
  #pragma unroll
  for (int v = 0; v < 8; ++v) {
    const int row = mt * 16 + v + 8 * hi;
    if (row < kC) Yb[(size_t)row * kH + nt * 16 + n] = acc[v] * scale;
  }
}

// ---------- Stage 3: C[b] = softmax(Y[b]*H[b]^T) * H[b]  (flash style) ------
constexpr int kCB = 64;          // c rows per block = 4 waves x 16
constexpr int kUT = 16;          // u tile (rows of H per TDM descriptor)
constexpr int kNT = (kT + kUT - 1) / kUT;   // 87 u tiles
constexpr int LDR = kH + 2;      // Y LDS row stride (194 dw) -> conflict-free
constexpr int LDS_S = 20;        // S-buffer row stride (dw)
// H buffer: TDM pads 2 dwords after every 64 dwords -> row stride 198 dw.
// 198 mod 64 = 6 -> 16-lane strided H^T fragment reads hit distinct banks;
// pads are even and never split a (k,k+1) pair, so float2 reads stay aligned.
constexpr int HROW = kH + 2 * (kH / 64);         // 198 dwords
constexpr int HBUF_DW = kUT * HROW;              // 3168 dwords per buffer

__device__ __forceinline__ int hidx(int r, int k) {
  return r * HROW + k + 2 * (k >> 6);
}

#if HAVE_TDM
// Issue a 2D TDM load: kUT x kH fp32 tile of H (row stride kH) -> padded LDS.
__device__ __forceinline__ void tdm_issue(const float* gsrc, unsigned lds_byte,
                                          int rows_remaining) {
  const unsigned long long ga = (unsigned long long)(uintptr_t)gsrc;
  v4u g0 = {1u,                                   // count=1 valid descriptor
            lds_byte,                             // lds_addr (bytes)
            (unsigned)ga,                         // global_addr[31:0]
            (unsigned)((ga >> 32) & 0x01FFFFFFu) | (2u << 30)}; // addr[56:32] | type=2
  v8i g1 = {(int)((2u << 16) | (1u << 20) | (5u << 22) | (1u << 25)),
            //    data=4B       pad_en      intvl=64dw     amount=2dw
            (int)((unsigned)kH << 16),            // tensor_dim0 = 192
            (int)(((unsigned)rows_remaining & 0xFFFFu) << 16), // tensor_dim1 lo
            (int)((((unsigned)rows_remaining >> 16) & 0xFFFFu) |
                  ((unsigned)kH << 16)),          // tensor_dim1 hi | tile_dim0
            kUT,                                  // tile_dim1=16, tile_dim2=0
            kH,                                   // tensor_dim0_stride = 192
            0, 0};
  v4i z = {};
#if __clang_major__ >= 23
  v8i z8 = {};
  __builtin_amdgcn_tensor_load_to_lds(g0, g1, z, z, z8, 0);
#else
  __builtin_amdgcn_tensor_load_to_lds(g0, g1, z, z, 0);
#endif
}
#endif

__global__ __launch_bounds__(128) void k_attn(const float* __restrict__ Y,
                                              const float* __restrict__ Hm,
                                              float* __restrict__ Out) {
  __shared__ float Ylds[kCB][LDR];
  __shared__ float Hbuf[2][HBUF_DW];
  __shared__ float Slds[4][16 * LDS_S];

  const int b = blockIdx.y;
  const int c0 = blockIdx.x * kCB;
  const int tid = threadIdx.x;
  const int wave = tid >> 5, lane = tid & 31;
  const int hi = lane >> 4, n = lane & 15;

  const float* __restrict__ Yb = Y + (size_t)b * kC * kH;
  const float* __restrict__ Hb = Hm + (size_t)b * kT * kH;

  // one-shot cooperative load of this block's 64 Y rows (queries), zero-padded
  for (int idx = tid; idx < kCB * (kH / 2); idx += 128) {
    const int r = idx / (kH / 2), c2 = idx % (kH / 2);
    const int row = c0 + r;
    v2f val; val.x = 0.f; val.y = 0.f;
    if (row < kC) val = *(const v2f*)(Yb + (size_t)row * kH + 2 * c2);
    Ylds[r][2 * c2] = val.x;
    Ylds[r][2 * c2 + 1] = val.y;
  }

  v8f O[12];
  #pragma unroll
  for (int t = 0; t < 12; ++t) O[t] = (v8f){};
  float mrun[8], srun[8];
  #pragma unroll
  for (int v = 0; v < 8; ++v) { mrun[v] = -3.0e38f; srun[v] = 0.f; }

  const float* __restrict__ yrow = &Ylds[wave * 16 + n][0];
  float* __restrict__ sb = &Slds[wave][0];

#if HAVE_TDM
  if (wave == 0) tdm_issue(Hb, (unsigned)(uintptr_t)&Hbuf[0][0], kT);
#endif

  for (int i = 0; i < kNT; ++i) {
    const int u0 = i * kUT;
#if HAVE_TDM
    // prefetch next tile into the other buffer, then wait for current tile
    if (wave == 0) {
      if (i + 1 < kNT) {
        tdm_issue(Hb + (size_t)(u0 + kUT) * kH,
                  (unsigned)(uintptr_t)&Hbuf[(i + 1) & 1][0], kT - (u0 + kUT));
        __builtin_amdgcn_s_wait_tensorcnt(1);
      } else {
        __builtin_amdgcn_s_wait_tensorcnt(0);
      }
    }
    __syncthreads();                       // publish tile i to all waves
    const float* __restrict__ hb = &Hbuf[i & 1][0];
#else
    __syncthreads();                       // previous tile fully consumed
    for (int idx = tid; idx < kUT * (kH / 2); idx += 128) {
      const int r = idx / (kH / 2), c2 = idx % (kH / 2);
      const int u = u0 + r;
      v2f val; val.x = 0.f; val.y = 0.f;
      if (u < kT) val = *(const v2f*)(Hb + (size_t)u * kH + 2 * c2);
      *(v2f*)&Hbuf[0][hidx(r, 2 * c2)] = val;
    }
    __syncthreads();
    const float* __restrict__ hb = &Hbuf[0][0];
#endif

    // S(16x16) = Ytile . Htile^T, K = 192
    v8f S = {};
    #pragma unroll 8
    for (int k0 = 0; k0 < kH; k0 += 4) {
      const int k = k0 + 2 * hi;
      v2f a = *(const v2f*)(yrow + k);                 // A: lane = c-row
      v2f bb = *(const v2f*)(hb + hidx(n, k));         // B = H^T: lane = u-col
      S = wmma_k4(a, bb, S);
    }
    if (u0 + n >= kT) {                                // mask tail u columns
      #pragma unroll
      for (int v = 0; v < 8; ++v) S[v] = -3.0e38f;
    }

    // online softmax; D-layout row v+8*hi spans the 16 lanes of this half
    float corr[8];
    #pragma unroll
    for (int v = 0; v < 8; ++v) {
      float r = S[v];
      r = fmaxf(r, __shfl_xor(r, 1, 32));
      r = fmaxf(r, __shfl_xor(r, 2, 32));
      r = fmaxf(r, __shfl_xor(r, 4, 32));
      r = fmaxf(r, __shfl_xor(r, 8, 32));
      const float mnew = fmaxf(mrun[v], r);
      corr[v] = __expf(mrun[v] - mnew);
      mrun[v] = mnew;
      const float p = __expf(S[v] - mnew);
      S[v] = p;
      float rs = p;
      rs += __shfl_xor(rs, 1, 32);
      rs += __shfl_xor(rs, 2, 32);
      rs += __shfl_xor(rs, 4, 32);
      rs += __shfl_xor(rs, 8, 32);
      srun[v] = srun[v] * corr[v] + rs;
    }
    #pragma unroll
    for (int t = 0; t < 12; ++t)
      #pragma unroll
      for (int v = 0; v < 8; ++v) O[t][v] *= corr[v];

    // transpose P: D-layout -> A-fragments via per-wave LDS bounce
    #pragma unroll
    for (int v = 0; v < 8; ++v) sb[(v + 8 * hi) * LDS_S + n] = S[v];
    v2f pa[4];
    #pragma unroll
    for (int kk = 0; kk < 4; ++kk)
      pa[kk] = *(const v2f*)(sb + n * LDS_S + 4 * kk + 2 * hi);

    // O += P(16x16) . Htile(16x192)
    #pragma unroll
    for (int t = 0; t < 12; ++t) {
      const int col = t * 16 + n;
      const int cadj = col + 2 * (col >> 6);
      v8f acc = O[t];
      #pragma unroll
      for (int kk = 0; kk < 4; ++kk) {
        const int k = 4 * kk + 2 * hi;
        v2f bb;
        bb.x = hb[k * HROW + cadj];
        bb.y = hb[(k + 1) * HROW + cadj];
        acc = wmma_k4(pa[kk], bb, acc);
      }
      O[t] = acc;
    }
#if HAVE_TDM
    __syncthreads();   // all waves done with Hbuf before it is re-targeted
#endif
  }

  // epilogue: normalize and store C[b, row, :]
  #pragma unroll
  for (int v = 0; v < 8; ++v) {
    const int row = c0 + wave * 16 + v + 8 * hi;
    if (row < kC) {
      const float inv = 1.0f / srun[v];
      #pragma unroll
      for (int t = 0; t < 12; ++t)
        Out[(size_t)b * kC * kH + (size_t)row * kH + t * 16 + n] = O[t][v] * inv;
    }
  }
}

extern "C" void kernel_launch(void* const* d_in, const int* in_sizes, int n_in,
                              void* d_out, int out_size, void* d_ws, size_t ws_size,
                              hipStream_t stream) {
  const float* X  = (const float*)d_in[0];   // [64,1380,96]
  const float* H  = (const float*)d_in[1];   // [64,1380,192]
  const float* W1 = (const float*)d_in[2];   // [96,192]
  const float* W2 = (const float*)d_in[3];   // [345,1380]
  float* out = (float*)d_out;                // [64,345,192]

  float* Z = (float*)d_ws;                   // [64,345,96]  ~8.5 MB
  float* Y = Z + (size_t)kB * kC * kX;       // [64,345,192] ~17 MB

  k_z   <<<dim3(kMT * (kX / 16), kB), 32,  0, stream>>>(W2, X, Z);
  k_y   <<<dim3(kMT * (kH / 16), kB), 32,  0, stream>>>(Z, W1, Y);
  k_attn<<<dim3((kC + kCB - 1) / kCB, kB), 128, 0, stream>>>(Y, H, out);
}